// TimeDropout_40853728920175
// MI455X (gfx1250) — compile-verified
//
#include <hip/hip_runtime.h>

// x: [B, T, D] f32. Zero a per-row time window [start, start+L) over all D
// channels when apply[b]. Pure streaming op: 128-bit NT loads/stores.

typedef __attribute__((ext_vector_type(4))) float v4f;

#define BATCH   16
#define TLEN    4096
#define DDIM    512
#define PROB    0.1f
#define MAXFRAC 0.1f

constexpr int VEC_PER_T      = DDIM / 4;                   // 128 vec4 per time step
constexpr int VEC_PER_ROW    = TLEN * VEC_PER_T;           // 524288
constexpr int THREADS        = 256;                        // 8 waves (wave32)
constexpr int VPT            = 8;                          // vec4 per thread
constexpr int CHUNK_VECS     = THREADS * VPT;              // 2048 vec4 per block
constexpr int CHUNKS_PER_ROW = VEC_PER_ROW / CHUNK_VECS;   // 256 blocks per row

__global__ __launch_bounds__(THREADS)
void time_dropout_kernel(const v4f* __restrict__ x,
                         const int*  __restrict__ lengths,
                         const float* __restrict__ u,
                         const float* __restrict__ s,
                         v4f* __restrict__ out) {
    const int b     = (int)blockIdx.x / CHUNKS_PER_ROW;   // row (block-uniform)
    const int chunk = (int)blockIdx.x % CHUNKS_PER_ROW;

    const size_t base = (size_t)b * VEC_PER_ROW
                      + (size_t)chunk * CHUNK_VECS
                      + threadIdx.x;

    // Issue all streaming (non-temporal) 128-bit loads up front; they do not
    // depend on the scalar row parameters, so they overlap the s_loads below.
    v4f v[VPT];
#pragma unroll
    for (int i = 0; i < VPT; ++i)
        v[i] = __builtin_nontemporal_load(&x[base + (size_t)i * THREADS]);

    // Block-uniform per-row dropout parameters (scalar path).
    // Matches reference: L = min(max(1, int(0.1*Lb)), max(Lb,1));
    //                    rng = max(Lb-L+1, 1); start = min(int(s*rng), rng-1)
    const int Lb = lengths[b];
    int L = (int)(MAXFRAC * (float)Lb);          // truncation, as astype(int32)
    L = (L < 1) ? 1 : L;
    const int cap = (Lb > 1) ? Lb : 1;
    L = (L > cap) ? cap : L;
    int rng = Lb - L + 1;
    rng = (rng < 1) ? 1 : rng;
    int start = (int)(s[b] * (float)rng);
    start = (start > rng - 1) ? (rng - 1) : start;
    const int zend   = start + L;
    const int apply  = (int)((Lb > 1) && (u[b] < PROB));

    const int vr0 = chunk * CHUNK_VECS + (int)threadIdx.x; // vec4 index in row
    const v4f zero = {0.f, 0.f, 0.f, 0.f};

#pragma unroll
    for (int i = 0; i < VPT; ++i) {
        const int t = (vr0 + i * THREADS) / VEC_PER_T;     // time index (>>7)
        const int z = apply & (int)(t >= start) & (int)(t < zend);
        v4f val = z ? zero : v[i];                          // v_cndmask x4
        __builtin_nontemporal_store(val, &out[base + (size_t)i * THREADS]);
    }
}

extern "C" void kernel_launch(void* const* d_in, const int* in_sizes, int n_in,
                              void* d_out, int out_size, void* d_ws, size_t ws_size,
                              hipStream_t stream) {
    (void)in_sizes; (void)n_in; (void)out_size; (void)d_ws; (void)ws_size;
    const v4f*   x       = (const v4f*)d_in[0];    // [16,4096,512] f32
    const int*   lengths = (const int*)d_in[1];    // [16] i32
    const float* u       = (const float*)d_in[2];  // [16] f32
    const float* sdraw   = (const float*)d_in[3];  // [16] f32
    v4f*         out     = (v4f*)d_out;

    dim3 grid(BATCH * CHUNKS_PER_ROW);             // 4096 blocks
    dim3 block(THREADS);                           // 256 threads = 8 waves
    time_dropout_kernel<<<grid, block, 0, stream>>>(x, lengths, u, sdraw, out);
}